// GraphGCN_88072599372182
// MI455X (gfx1250) — compile-verified
//
#include <hip/hip_runtime.h>
#include <hip/hip_bf16.h>

typedef __attribute__((ext_vector_type(2))) float v2f;
typedef __attribute__((ext_vector_type(8))) float v8f;

#define N_NODES   100000
#define N_EDGES   1600000
#define F_IN      128
#define F_HID     64
#define N_CLASSES 2
#define N_GRAPHS  128

// ---------------------------------------------------------------- utilities
__global__ void k_fill4(float4* __restrict__ p, float v, long n4) {
    long i = (long)blockIdx.x * blockDim.x + threadIdx.x;
    if (i < n4) p[i] = make_float4(v, v, v, v);
}

// deg[dst] += 1 for every real edge (deg pre-initialized to 1.0 for self loops)
__global__ void k_deg_edges(const long long* __restrict__ dst, float* __restrict__ deg) {
    int e = blockIdx.x * blockDim.x + threadIdx.x;
    if (e < N_EDGES) atomicAdd(&deg[(int)dst[e]], 1.0f);
}

__global__ void k_dis(const float* __restrict__ deg, float* __restrict__ dis) {
    int n = blockIdx.x * blockDim.x + threadIdx.x;
    if (n < N_NODES) dis[n] = rsqrtf(deg[n]);   // deg >= 1 always (self loop)
}

__global__ void k_norm(const long long* __restrict__ src, const long long* __restrict__ dst,
                       const float* __restrict__ dis, float* __restrict__ norm) {
    int e = blockIdx.x * blockDim.x + threadIdx.x;
    if (e < N_EDGES) norm[e] = dis[(int)src[e]] * dis[(int)dst[e]];
}

// ---------------------------------------------------------------- WMMA GEMM
// C[M x 64] = A[M x K] * B[K x 64], row-major f32, M % 32 == 0, K % 4 == 0.
// Block = 256 threads = 8 waves covering a 32x64 output tile (2 M-subtiles x
// 4 N-subtiles of 16x16), V_WMMA_F32_16X16X4_F32 per K-step, K fully unrolled
// at compile time so all LDS/global accesses use immediate offsets.
//
// Both operand tiles are staged through the CDNA5 async copy path
// (global_load_async_to_lds_b128 -> s_wait_asynccnt 0 -> barrier):
//   - B entire matrix (K*64*4 <= 32 KB), linear; LDS bank = n mod 64 ->
//     16 distinct banks per half-wave read, conflict-free.
//   - A 32-row tile with row stride padded to K+4 floats so lane r hits bank
//     (4r + k) mod 64 -> conflict-free (unpadded stride 512B/256B would put
//     all 16 rows on one bank). Pair (k, k+1) is contiguous -> ds_load_b64.
//
// Fragment layouts per CDNA5 ISA 7.12.2:
//   A 16x4 f32 : lanes 0-15 rows M=0..15; VGPR0 = K{0 | 2}, VGPR1 = K{1 | 3}
//   B 4x16 f32 : lane n holds column N=n; same K split across the two regs
//   C/D 16x16  : reg r -> row (r + 8*laneHalf), col = lane&15
template<int K>
__global__ void __launch_bounds__(256)
k_gemm_wmma_f32(const float* __restrict__ A, const float* __restrict__ B,
                float* __restrict__ C) {
    constexpr int N   = 64;
    constexpr int AST = K + 4;                    // padded A row stride (floats)
    __shared__ float Blds[K * N];
    __shared__ float Alds[32 * AST];

    const int tid   = threadIdx.x;
    const int m0blk = blockIdx.x * 32;

    // ---- async-stage B (linear, 128-bit granules) ----
#pragma unroll
    for (int i = tid; i < (K * N) / 4; i += 256) {
        unsigned loff = (unsigned)(size_t)(&Blds[i * 4]);
        const float4* gp = (const float4*)B + i;
        asm volatile("global_load_async_to_lds_b128 %0, %1, off"
                     :: "v"(loff), "v"(gp) : "memory");
    }
    // ---- async-stage A tile (row-padded) ----
#pragma unroll
    for (int i = tid; i < (32 * K) / 4; i += 256) {
        const int r = i / (K / 4);
        const int c = i % (K / 4);
        unsigned loff = (unsigned)(size_t)(&Alds[r * AST + c * 4]);
        const float4* gp = (const float4*)(A + (long)(m0blk + r) * K) + c;
        asm volatile("global_load_async_to_lds_b128 %0, %1, off"
                     :: "v"(loff), "v"(gp) : "memory");
    }
    asm volatile("s_wait_asynccnt 0" ::: "memory");
    __syncthreads();

    const int wave = tid >> 5;
    const int lane = tid & 31;
    const int half = lane >> 4;                   // 0 = lanes 0-15, 1 = 16-31
    const int l16  = lane & 15;
    const int msub = (wave >> 2) * 16;            // 0 or 16 within the block tile
    const int n0   = (wave & 3) * 16;

    v8f acc = {};
    const float* __restrict__ arow = &Alds[(msub + l16) * AST + 2 * half];
    const float* __restrict__ bcol = &Blds[2 * half * N + n0 + l16];
#pragma unroll
    for (int k0 = 0; k0 < K; k0 += 4) {
        v2f a, b;
        a.x = arow[k0 + 0];                       // ds_load_b64 @ imm offset
        a.y = arow[k0 + 1];
        b.x = bcol[(k0 + 0) * N];                 // ds_load_b32 @ imm offset
        b.y = bcol[(k0 + 1) * N];
        acc = __builtin_amdgcn_wmma_f32_16x16x4_f32(
            /*neg_a=*/false, a, /*neg_b=*/false, b,
            /*c_mod=*/(short)0, acc, /*reuse_a=*/false, /*reuse_b=*/false);
    }
    float* __restrict__ crow = C + (long)(m0blk + msub + 8 * half) * N + n0 + l16;
#pragma unroll
    for (int r = 0; r < 8; ++r)
        crow[r * N] = acc[r];
}

// ------------------------------------------------------- edge scatter-add
// out[dst] += h[src] * norm[e]; one thread per (edge, 4 features):
// one b128 gather + 4 f32 L2 atomics.
__global__ void k_aggregate(const long long* __restrict__ src, const long long* __restrict__ dst,
                            const float* __restrict__ norm,
                            const float* __restrict__ h, float* __restrict__ out) {
    long t = (long)blockIdx.x * blockDim.x + threadIdx.x;
    if (t >= (long)N_EDGES * (F_HID / 4)) return;
    int e  = (int)(t >> 4);
    int f  = (int)(t & 15) << 2;
    int s  = (int)src[e], d = (int)dst[e];
    float nm = norm[e];
    const float4 hv = *(const float4*)(h + (long)s * F_HID + f);
    float* o = out + (long)d * F_HID + f;
    atomicAdd(o + 0, hv.x * nm);
    atomicAdd(o + 1, hv.y * nm);
    atomicAdd(o + 2, hv.z * nm);
    atomicAdd(o + 3, hv.w * nm);
}

// agg = relu(agg + dis[n]^2 * h (self loop) + bias)   (in place, float4)
__global__ void k_bias_relu_self(float* __restrict__ agg, const float* __restrict__ h,
                                 const float* __restrict__ dis, const float* __restrict__ bias) {
    long t = (long)blockIdx.x * blockDim.x + threadIdx.x;
    if (t >= (long)N_NODES * (F_HID / 4)) return;
    int n = (int)(t >> 4);
    int f = (int)(t & 15) << 2;
    float di = dis[n];
    float sl = di * di;
    float4 av = ((float4*)agg)[t];
    float4 hv = ((const float4*)h)[t];
    float4 bv = *(const float4*)(bias + f);
    float4 r;
    r.x = fmaxf(av.x + sl * hv.x + bv.x, 0.0f);
    r.y = fmaxf(av.y + sl * hv.y + bv.y, 0.0f);
    r.z = fmaxf(av.z + sl * hv.z + bv.z, 0.0f);
    r.w = fmaxf(av.w + sl * hv.w + bv.w, 0.0f);
    ((float4*)agg)[t] = r;
}

// ---------------------------------------------------------------- pooling
__global__ void k_pool_counts(const long long* __restrict__ batch, float* __restrict__ cnt) {
    int n = blockIdx.x * blockDim.x + threadIdx.x;
    if (n < N_NODES) atomicAdd(&cnt[(int)batch[n]], 1.0f);
}

__global__ void k_pool_sums(const long long* __restrict__ batch, const float* __restrict__ h,
                            float* __restrict__ pooled) {
    long t = (long)blockIdx.x * blockDim.x + threadIdx.x;
    if (t >= (long)N_NODES * (F_HID / 4)) return;
    int n = (int)(t >> 4);
    int f = (int)(t & 15) << 2;
    int g = (int)batch[n];
    float4 hv = ((const float4*)h)[t];
    float* p = pooled + (long)g * F_HID + f;
    atomicAdd(p + 0, hv.x);
    atomicAdd(p + 1, hv.y);
    atomicAdd(p + 2, hv.z);
    atomicAdd(p + 3, hv.w);
}

// out[g, c] = (pooled[g,:]/max(cnt,1)) . Wf[:,c] + bf[c]   (128x2 outputs)
__global__ void k_final(const float* __restrict__ pooled, const float* __restrict__ cnt,
                        const float* __restrict__ Wf, const float* __restrict__ bf,
                        float* __restrict__ out) {
    int t = threadIdx.x;
    if (t >= N_GRAPHS * N_CLASSES) return;
    int g = t >> 1, c = t & 1;
    float inv = 1.0f / fmaxf(cnt[g], 1.0f);
    float acc = bf[c];
    for (int k = 0; k < F_HID; ++k)
        acc += pooled[g * F_HID + k] * inv * Wf[k * N_CLASSES + c];
    out[t] = acc;
}

// ---------------------------------------------------------------- launcher
extern "C" void kernel_launch(void* const* d_in, const int* in_sizes, int n_in,
                              void* d_out, int out_size, void* d_ws, size_t ws_size,
                              hipStream_t stream) {
    (void)in_sizes; (void)n_in; (void)out_size; (void)ws_size;

    const float*     x     = (const float*)d_in[0];
    const long long* edge  = (const long long*)d_in[1];   // [2, E] int64
    const long long* srcp  = edge;
    const long long* dstp  = edge + N_EDGES;
    const long long* batch = (const long long*)d_in[2];   // [N] int64, sorted
    const float*     W1    = (const float*)d_in[3];       // [128, 64]
    const float*     b1    = (const float*)d_in[4];
    const float*     W2    = (const float*)d_in[5];       // [64, 64]
    const float*     b2    = (const float*)d_in[6];
    const float*     Wf    = (const float*)d_in[7];       // [64, 2]
    const float*     bf    = (const float*)d_in[8];
    float*           out   = (float*)d_out;

    // workspace carve-up (floats)
    float* ws     = (float*)d_ws;
    float* bufA   = ws;                                     // N_NODES*64
    float* bufB   = bufA + (size_t)N_NODES * F_HID;         // N_NODES*64
    float* deg    = bufB + (size_t)N_NODES * F_HID;         // N_NODES
    float* dis    = deg  + N_NODES;                         // N_NODES
    float* norm   = dis  + N_NODES;                         // N_EDGES
    float* pooled = norm + N_EDGES;                         // 128*64
    float* cnt    = pooled + N_GRAPHS * F_HID;              // 128

    const int TB = 256;
    const int gNodes = (N_NODES + TB - 1) / TB;
    const int gEdges = (N_EDGES + TB - 1) / TB;
    const long nodeF4 = (long)N_NODES * (F_HID / 4);        // 1.6M float4
    const long edgeF4 = (long)N_EDGES * (F_HID / 4);        // 25.6M threads
    const int gNodeF4 = (int)((nodeF4 + TB - 1) / TB);
    const int gEdgeF4 = (int)((edgeF4 + TB - 1) / TB);

    // --- normalization coefficients -------------------------------------
    k_fill4<<<(N_NODES / 4 + TB - 1) / TB, TB, 0, stream>>>((float4*)deg, 1.0f, N_NODES / 4);
    k_deg_edges<<<gEdges, TB, 0, stream>>>(dstp, deg);
    k_dis<<<gNodes, TB, 0, stream>>>(deg, dis);
    k_norm<<<gEdges, TB, 0, stream>>>(srcp, dstp, dis, norm);

    // --- layer 1: h0 = x @ W1 ; agg ; relu(+self loop, +b1) -------------
    k_gemm_wmma_f32<F_IN><<<N_NODES / 32, 256, 0, stream>>>(x, W1, bufA);
    k_fill4<<<gNodeF4, TB, 0, stream>>>((float4*)bufB, 0.0f, nodeF4);
    k_aggregate<<<gEdgeF4, TB, 0, stream>>>(srcp, dstp, norm, bufA, bufB);
    k_bias_relu_self<<<gNodeF4, TB, 0, stream>>>(bufB, bufA, dis, b1);   // bufB = h1

    // --- layer 2: h1 @ W2 ; agg ; relu(+self loop, +b2) -----------------
    k_gemm_wmma_f32<F_HID><<<N_NODES / 32, 256, 0, stream>>>(bufB, W2, bufA);
    k_fill4<<<gNodeF4, TB, 0, stream>>>((float4*)bufB, 0.0f, nodeF4);    // h1 dead, reuse
    k_aggregate<<<gEdgeF4, TB, 0, stream>>>(srcp, dstp, norm, bufA, bufB);
    k_bias_relu_self<<<gNodeF4, TB, 0, stream>>>(bufB, bufA, dis, b2);   // bufB = h2

    // --- global mean pool + head ----------------------------------------
    k_fill4<<<1, TB, 0, stream>>>((float4*)pooled, 0.0f, (N_GRAPHS * F_HID) / 4);
    k_fill4<<<1, TB, 0, stream>>>((float4*)cnt, 0.0f, N_GRAPHS / 4);
    k_pool_counts<<<gNodes, TB, 0, stream>>>(batch, cnt);
    k_pool_sums<<<gNodeF4, TB, 0, stream>>>(batch, bufB, pooled);
    k_final<<<1, 256, 0, stream>>>(pooled, cnt, Wf, bf, out);
}